// PowerSpectrum_56126632624554
// MI455X (gfx1250) — compile-verified
//
#include <hip/hip_runtime.h>
#include <math.h>

// CDNA5 fp32 WMMA operand types: A/B 16x4 / 4x16 fp32 = 2 VGPRs, C/D 16x16 f32 = 8 VGPRs
typedef __attribute__((ext_vector_type(2))) float v2f;
typedef __attribute__((ext_vector_type(4))) float v4f;
typedef __attribute__((ext_vector_type(8))) float v8f;

#define PS_WAVES 4
#define PS_NFEAT 3696   // 7 * 528
#define PS_NPART 528    // 32 diag + 496 upper
#define PS_STRIDE 3700  // padded row stride: slot 3696 = dump word, 16B-aligned rows
#define PS_INV_SQRT2 0.70710678118654752f

__device__ __forceinline__ v8f wmma4(v2f a, v2f b, v8f c) {
  // V_WMMA_F32_16X16X4_F32: (neg_a, A, neg_b, B, c_mod, C, reuse_a, reuse_b)
  return __builtin_amdgcn_wmma_f32_16x16x4_f32(false, a, false, b, (short)0, c,
                                               false, false);
}

// Load one pre-scaled operand element. o = k0 + c is compile-time after
// unrolling: dead branches fold, so no OOB loads are ever emitted.
//  - o+2 <  M : valid for both hi-halves -> load from hi-shifted row base
//  - o   <  M : valid only for hi==0     -> load from un-shifted base (always
//               in-bounds), cndmask hi lanes to zero
//  - else     : constant zero, no load
template <int M>
__device__ __forceinline__ float ldelt(const float* pl, const float* prow, int hi,
                                       int o, float s) {
  if (o + 2 < M) {
    return pl[o] * s;
  } else if (o < M) {
    const float x = prow[o] * s;
    return hi ? 0.f : x;
  } else {
    return 0.f;
  }
}

// Diagonal-block tile (acc00: base=0, acc11: base=16). Per lane the elements
// are ordered upper (v<tt), diagonal (v==tt), lower (v>tt); every element does
// one unconditional ds_store with a cndmask-selected byte offset (upper slot /
// diag slot / dump word). Values are stored raw (= sqrt2*scale*G); the /sqrt2
// diag fixup happens at writeout. Returns raw Frobenius partial (x2 scaled).
__device__ __forceinline__ float emit_diag_tile(v8f t, int base, int lane,
                                                float* psw, int dumpB) {
  const int n = base + (lane & 15);
  const int mh = base + ((lane >> 4) << 3);
  int uoff = (32 + 31 * mh - (mh * (mh - 1)) / 2 + (n - mh - 1)) * 4;
  const int nB = n * 4;   // diag feature slot (index m == n), bytes
  const int tt = n - mh;  // diagonal element position within this lane's column
  float q = 0.f;
#pragma unroll
  for (int v = 0; v < 8; ++v) {
    const float val = t[v];
    q += val * val;
    const int inner = (v == tt) ? nB : dumpB;
    const int idx = (v < tt) ? uoff : inner;
    *(float*)((char*)psw + idx) = val;
    uoff += (30 - mh - v) * 4;  // uidx(m+1)-uidx(m) = 30-m, in bytes
  }
  return q;
}

// Off-diagonal tile acc01 (rows 0-15 x cols 16-31): every element is an upper
// pair -> unconditional store of the raw value, no compares, no muls.
__device__ __forceinline__ float emit_off_tile(v8f t, int lane, float* psw) {
  const int n = 16 + (lane & 15);
  const int mh = (lane >> 4) << 3;
  int uoff = (32 + 31 * mh - (mh * (mh - 1)) / 2 + (n - mh - 1)) * 4;
  float q = 0.f;
#pragma unroll
  for (int v = 0; v < 8; ++v) {
    const float val = t[v];
    q += val * val;
    *(float*)((char*)psw + uoff) = val;
    uoff += (30 - mh - v) * 4;
  }
  return q;
}

template <int M>
__device__ __forceinline__ void process_l(const float* __restrict__ se, size_t j,
                                          int lane, float* psw, int dumpB,
                                          float s, float& qd, float& qo) {
  const int r = lane & 15;
  const int hi = lane >> 4;
  const float* prow0 = se + (j * 32 + (size_t)r) * M;  // row r (contiguous M floats)
  const float* prow1 = prow0 + (size_t)16 * M;         // row r+16
  const float* plo = prow0 + 2 * hi;                   // hi-half feeds K+2..K+3
  const float* phi = prow1 + 2 * hi;

  v8f a00 = {}, a01 = {}, a11 = {};
  constexpr int KS = (M + 3) / 4;
#pragma unroll
  for (int kk = 0; kk < KS; ++kk) {
    const int k0 = 4 * kk;
    // f32 WMMA A-tile packing == B-tile packing for a Gram product (B = S^T).
    v2f lo, hv;
    lo.x = ldelt<M>(plo, prow0, hi, k0 + 0, s);
    lo.y = ldelt<M>(plo, prow0, hi, k0 + 1, s);
    hv.x = ldelt<M>(phi, prow1, hi, k0 + 0, s);
    hv.y = ldelt<M>(phi, prow1, hi, k0 + 1, s);
    a00 = wmma4(lo, lo, a00);  // rows 0-15  x rows 0-15
    a01 = wmma4(lo, hv, a01);  // rows 0-15  x rows 16-31 (all upper pairs)
    a11 = wmma4(hv, hv, a11);  // rows 16-31 x rows 16-31
  }

  qd += emit_diag_tile(a00, 0, lane, psw, dumpB);
  qo += emit_off_tile(a01, lane, psw);
  qd += emit_diag_tile(a11, 16, lane, psw, dumpB);
}

__global__ void __launch_bounds__(PS_WAVES * 32)
powerspectrum_kernel(const float* __restrict__ se0, const float* __restrict__ se1,
                     const float* __restrict__ se2, const float* __restrict__ se3,
                     const float* __restrict__ se4, const float* __restrict__ se5,
                     const float* __restrict__ se6, float* __restrict__ out, int J) {
  __shared__ float ps[PS_WAVES][PS_STRIDE];  // 4 * 14800 B = 59200 B
  const int lane = threadIdx.x & 31;
  const int w = threadIdx.x >> 5;
  size_t j = (size_t)blockIdx.x * PS_WAVES + (size_t)w;
  if (j >= (size_t)J) j = (size_t)J - 1;  // wave-uniform clamp: EXEC stays all-1s
  float* psw = &ps[w][0];

  // Row pre-scale s_l = sqrt(sqrt(2)/sqrt(2l+1)) folds both the 1/sqrt(2l+1)
  // Gram scale and the sqrt(2) upper-feature factor into the inputs.
  float qd = 0.f, qo = 0.f;  // raw diag-tile / off-tile Frobenius partials
  process_l<1>(se0, j, lane, psw + 0 * PS_NPART, (PS_NFEAT - 0 * PS_NPART) * 4,
               1.18920711500272107f, qd, qo);
  process_l<3>(se1, j, lane, psw + 1 * PS_NPART, (PS_NFEAT - 1 * PS_NPART) * 4,
               0.90360200360984493f, qd, qo);
  process_l<5>(se2, j, lane, psw + 2 * PS_NPART, (PS_NFEAT - 2 * PS_NPART) * 4,
               0.79527072876705059f, qd, qo);
  process_l<7>(se3, j, lane, psw + 3 * PS_NPART, (PS_NFEAT - 3 * PS_NPART) * 4,
               0.73111044591965710f, qd, qo);
  process_l<9>(se4, j, lane, psw + 4 * PS_NPART, (PS_NFEAT - 4 * PS_NPART) * 4,
               0.68658904796903922f, qd, qo);
  process_l<11>(se5, j, lane, psw + 5 * PS_NPART, (PS_NFEAT - 5 * PS_NPART) * 4,
                0.65299420575648443f, qd, qo);
  process_l<13>(se6, j, lane, psw + 6 * PS_NPART, (PS_NFEAT - 6 * PS_NPART) * 4,
                0.62628449716251850f, qd, qo);

  // With val = sqrt2*scale*G: ||ps||^2 = 0.5*sum(diag tiles val^2) + sum(off val^2)
  float p = qo + 0.5f * qd;
#pragma unroll
  for (int off = 16; off > 0; off >>= 1) p += __shfl_xor(p, off, 32);
  const float rn = 1.0f / fmaxf(sqrtf(p), 1e-12f);
  const float rnD = rn * PS_INV_SQRT2;  // diag features carry an extra sqrt2

  __syncthreads();  // order LDS scatter vs. the vector re-read below

  // Coalesced normalized writeout: 924 float4 per j (b128 LDS loads + stores).
  // Diag features are the first 32 of each 528-block -> float4 index (f4 % 132) < 8.
  const v4f* src = (const v4f*)psw;               // wave row is 16B-aligned
  v4f* dst = (v4f*)(out + j * (size_t)PS_NFEAT);  // 3696*4 B = 16B multiple
  for (int f4 = lane; f4 < PS_NFEAT / 4; f4 += 32) {
    const v4f x = src[f4];
    const float m = ((f4 % 132) < 8) ? rnD : rn;
    dst[f4] = x * m;
  }
}

extern "C" void kernel_launch(void* const* d_in, const int* in_sizes, int n_in,
                              void* d_out, int out_size, void* d_ws, size_t ws_size,
                              hipStream_t stream) {
  const float* se0 = (const float*)d_in[0];
  const float* se1 = (const float*)d_in[1];
  const float* se2 = (const float*)d_in[2];
  const float* se3 = (const float*)d_in[3];
  const float* se4 = (const float*)d_in[4];
  const float* se5 = (const float*)d_in[5];
  const float* se6 = (const float*)d_in[6];
  float* out = (float*)d_out;

  const int J = in_sizes[0] / 32;  // se_0 is (J, 4, 8, 1)
  const int blocks = (J + PS_WAVES - 1) / PS_WAVES;
  powerspectrum_kernel<<<blocks, PS_WAVES * 32, 0, stream>>>(se0, se1, se2, se3,
                                                             se4, se5, se6, out, J);
}